// GIN_89627377533181
// MI455X (gfx1250) — compile-verified
//
#include <hip/hip_runtime.h>
#include <hip/hip_bf16.h>

#define H 128
#define LDSPAD 132   // 128 + 4 pad: row stride mod 64 banks = 4 -> conflict-free column reads

typedef float v2f __attribute__((ext_vector_type(2)));
typedef float v8f __attribute__((ext_vector_type(8)));
typedef int   v4i __attribute__((ext_vector_type(4)));

#if __has_builtin(__builtin_amdgcn_global_load_async_to_lds_b128) && \
    __has_builtin(__builtin_amdgcn_s_wait_asynccnt)
#define HAVE_ASYNC_LDS 1
typedef __attribute__((address_space(1))) v4i* gv4i_p;
typedef __attribute__((address_space(3))) v4i* lv4i_p;
#else
#define HAVE_ASYNC_LDS 0
#endif

// ---------------------------------------------------------------- utilities
__global__ void zero_kernel(float* __restrict__ p, int n) {
    int gid = blockIdx.x * blockDim.x + threadIdx.x;
    if (gid < n) p[gid] = 0.0f;
}

// ------------------------------------------------------- edge scatter (agg += h[src])
// one thread per (edge, 4-channel chunk): 32 chunks per edge
__global__ __launch_bounds__(256) void scatter_add_kernel(
    const float* __restrict__ h,
    const long long* __restrict__ src,
    const long long* __restrict__ dst,
    float* __restrict__ agg,
    int n_edges)
{
    unsigned int gid = blockIdx.x * 256u + threadIdx.x;
    unsigned int total = (unsigned int)n_edges * 32u;
    if (gid >= total) return;
    int e  = gid >> 5;
    int c  = (gid & 31) << 2;
    int s  = (int)src[e];
    int d  = (int)dst[e];
    const float4 v = *(const float4*)(h + (size_t)s * H + c);
    float* p = agg + (size_t)d * H + c;
    atomicAdd(p + 0, v.x);
    atomicAdd(p + 1, v.y);
    atomicAdd(p + 2, v.z);
    atomicAdd(p + 3, v.w);
}

// ------------------------------------------------------- fused 2-layer MLP + BN stats
// Persistent: block = 256 threads = 8 wave32; each wave preloads its W1/W2 column-stripe
// fragments into registers once, then grid-strides over 16-row tiles.
__global__ __launch_bounds__(256) void mlp_kernel(
    const float* __restrict__ agg,
    const float* __restrict__ W1, const float* __restrict__ b1,
    const float* __restrict__ W2, const float* __restrict__ b2,
    float* __restrict__ z,
    float* __restrict__ gsum, float* __restrict__ gsumsq,
    int n_nodes)
{
    __shared__ float Alds[16 * LDSPAD];
    __shared__ float Zlds[16 * LDSPAD];
    __shared__ float lsum[H];
    __shared__ float lsq[H];

    const int tid  = threadIdx.x;
    const int wave = tid >> 5;
    const int lane = tid & 31;
    const int m    = lane & 15;        // A row / C,D col-within-tile
    const int g    = lane >> 4;        // lane half-group
    const int col  = (wave << 4) | m;  // this lane's output column (0..127)

    // ---- one-time: preload B fragments (column stripe of W1/W2) into registers
    // A 16x4 f32 frag layout => B 4x16: vgpr0 = K=k0+2g, vgpr1 = K=k0+2g+1
    v2f B1f[32], B2f[32];
    #pragma unroll
    for (int t = 0; t < 32; ++t) {
        const int k0 = t * 4;
        B1f[t].x = W1[(k0 + 2 * g) * H + col];
        B1f[t].y = W1[(k0 + 2 * g + 1) * H + col];
        B2f[t].x = W2[(k0 + 2 * g) * H + col];
        B2f[t].y = W2[(k0 + 2 * g + 1) * H + col];
    }
    const float bias1 = b1[col];
    const float bias2 = b2[col];

    if (tid < H) { lsum[tid] = 0.0f; lsq[tid] = 0.0f; }
    __syncthreads();

    float psum = 0.0f, psq = 0.0f;   // BN partials, accumulated across tiles
    const int ntiles = (n_nodes + 15) >> 4;

    for (int tile = blockIdx.x; tile < ntiles; tile += gridDim.x) {
        const int row0 = tile << 4;

        // stage A tile (16 x 128) into LDS; 512 float4 chunks over 256 threads
        #pragma unroll
        for (int it = 0; it < 2; ++it) {
            const int idx = tid + it * 256;
            const int r  = idx >> 5;
            const int c4 = (idx & 31) << 2;
            int rr = row0 + r; if (rr >= n_nodes) rr = n_nodes - 1;  // clamp (N%16==0 normally)
            const float* gp = agg + (size_t)rr * H + c4;
#if HAVE_ASYNC_LDS
            __builtin_amdgcn_global_load_async_to_lds_b128(
                (gv4i_p)gp, (lv4i_p)&Alds[r * LDSPAD + c4], 0, 0);
#else
            *(float4*)(&Alds[r * LDSPAD + c4]) = *(const float4*)gp;
#endif
        }
#if HAVE_ASYNC_LDS
        __builtin_amdgcn_s_wait_asynccnt(0);
#endif
        __syncthreads();

        // ---- GEMM1: Z1 = relu(A @ W1 + b1) -> LDS
        v8f acc = {};
        const float* Arow = &Alds[m * LDSPAD];
        #pragma unroll
        for (int t = 0; t < 32; ++t) {
            v2f a;
            a.x = Arow[4 * t + 2 * g];
            a.y = Arow[4 * t + 2 * g + 1];
            acc = __builtin_amdgcn_wmma_f32_16x16x4_f32(
                false, a, false, B1f[t], (short)0, acc, false, false);
        }
        #pragma unroll
        for (int r = 0; r < 8; ++r) {
            float v = acc[r] + bias1;
            v = v > 0.0f ? v : 0.0f;
            Zlds[(r + 8 * g) * LDSPAD + col] = v;
        }
        __syncthreads();

        // ---- GEMM2: Z = Z1 @ W2 + b2 -> global; accumulate BN partials in regs
        v8f acc2 = {};
        const float* Zrow = &Zlds[m * LDSPAD];
        #pragma unroll
        for (int t = 0; t < 32; ++t) {
            v2f a;
            a.x = Zrow[4 * t + 2 * g];
            a.y = Zrow[4 * t + 2 * g + 1];
            acc2 = __builtin_amdgcn_wmma_f32_16x16x4_f32(
                false, a, false, B2f[t], (short)0, acc2, false, false);
        }
        #pragma unroll
        for (int r = 0; r < 8; ++r) {
            float v = acc2[r] + bias2;
            const int row = row0 + r + 8 * g;
            if (row < n_nodes) {
                z[(size_t)row * H + col] = v;
                psum += v;
                psq  += v * v;
            }
        }
        __syncthreads();   // Zlds/Alds safe to overwrite next tile
    }

    // block-level BN reduction: lane partials -> LDS -> one global atomic per channel
    atomicAdd(&lsum[col], psum);
    atomicAdd(&lsq[col],  psq);
    __syncthreads();
    if (tid < H) {
        atomicAdd(&gsum[tid],   lsum[tid]);
        atomicAdd(&gsumsq[tid], lsq[tid]);
    }
}

// ------------------------------------------------------- BN fold + apply
__global__ void bn_finalize_kernel(
    const float* __restrict__ gsum, const float* __restrict__ gsumsq,
    const float* __restrict__ gamma, const float* __restrict__ beta,
    float* __restrict__ scale, float* __restrict__ shift, float inv_n)
{
    int c = threadIdx.x;
    float mu  = gsum[c] * inv_n;
    float var = gsumsq[c] * inv_n - mu * mu;
    float s   = gamma[c] * rsqrtf(var + 1e-5f);
    scale[c] = s;
    shift[c] = beta[c] - mu * s;
}

__global__ __launch_bounds__(256) void bn_apply_kernel(
    const float* __restrict__ z,
    const float* __restrict__ scale, const float* __restrict__ shift,
    float* __restrict__ h, int total4)
{
    int gid = blockIdx.x * 256 + threadIdx.x;
    if (gid >= total4) return;
    int c4 = (gid & 31) << 2;
    const float4 v = *(const float4*)(z + (size_t)gid * 4);
    float4 o;
    o.x = fmaxf(v.x * scale[c4 + 0] + shift[c4 + 0], 0.0f);
    o.y = fmaxf(v.y * scale[c4 + 1] + shift[c4 + 1], 0.0f);
    o.z = fmaxf(v.z * scale[c4 + 2] + shift[c4 + 2], 0.0f);
    o.w = fmaxf(v.w * scale[c4 + 3] + shift[c4 + 3], 0.0f);
    *(float4*)(h + (size_t)gid * 4) = o;
}

// ------------------------------------------------------- global mean pool
__global__ __launch_bounds__(256) void count_kernel(
    const long long* __restrict__ batch, float* __restrict__ counts, int n_nodes)
{
    int gid = blockIdx.x * 256 + threadIdx.x;
    if (gid >= n_nodes) return;
    atomicAdd(&counts[(int)batch[gid]], 1.0f);
}

__global__ __launch_bounds__(256) void pool_sum_kernel(
    const float* __restrict__ h, const long long* __restrict__ batch,
    float* __restrict__ gout, int total4)
{
    int gid = blockIdx.x * 256 + threadIdx.x;
    if (gid >= total4) return;
    int node = gid >> 5;
    int c4   = (gid & 31) << 2;
    int gidx = (int)batch[node];
    const float4 v = *(const float4*)(h + (size_t)node * H + c4);
    float* p = gout + (size_t)gidx * H + c4;
    atomicAdd(p + 0, v.x);
    atomicAdd(p + 1, v.y);
    atomicAdd(p + 2, v.z);
    atomicAdd(p + 3, v.w);
}

__global__ void pool_div_kernel(
    float* __restrict__ gout, const float* __restrict__ counts, int n)
{
    int gid = blockIdx.x * blockDim.x + threadIdx.x;
    if (gid >= n) return;
    float c = counts[gid >> 7];          // graph id = gid / 128
    gout[gid] = gout[gid] / fmaxf(c, 1.0f);
}

// ---------------------------------------------------------------- launcher
extern "C" void kernel_launch(void* const* d_in, const int* in_sizes, int n_in,
                              void* d_out, int out_size, void* d_ws, size_t ws_size,
                              hipStream_t stream) {
    const float*     x    = (const float*)d_in[0];
    const long long* ei   = (const long long*)d_in[1];  // (2, E) int64
    const long long* bat  = (const long long*)d_in[2];  // (N,)  int64
    const float*     W1   = (const float*)d_in[3];      // (3,128,128)
    const float*     b1   = (const float*)d_in[4];      // (3,128)
    const float*     W2   = (const float*)d_in[5];
    const float*     b2   = (const float*)d_in[6];
    const float*     gam  = (const float*)d_in[7];
    const float*     bet  = (const float*)d_in[8];

    const int n_nodes  = in_sizes[0] / H;               // 100000
    const int n_edges  = (int)(in_sizes[1] / 2);        // 1600000
    const int n_layers = in_sizes[3] / (H * H);         // 3
    const size_t NH = (size_t)n_nodes * H;

    const long long* src = ei;
    const long long* dst = ei + n_edges;

    float* ws     = (float*)d_ws;
    float* agg    = ws;                 // N*H
    float* zbuf   = ws + NH;            // N*H
    float* stats  = ws + 2 * NH;        // sum(128) sumsq(128) scale(128) shift(128)
    float* counts = stats + 512;        // 64

    float* hbuf = (float*)d_out;        // node embeddings region (N*H)
    float* gout = hbuf + NH;            // graph embeddings region (64*H)

    const int edge_blocks = (int)(((size_t)n_edges * 32 + 255) / 256);
    const int feat4       = n_nodes * 32;               // N*H/4
    const int feat_blocks = (feat4 + 255) / 256;
    const int ntiles      = (n_nodes + 15) / 16;        // 6250
    int mlp_grid = 625;                                  // persistent: ~10 tiles/block
    if (mlp_grid > ntiles) mlp_grid = ntiles;

    const float* hcur = x;
    for (int l = 0; l < n_layers; ++l) {
        // agg = h  (self term), then scatter-add neighbor features
        (void)hipMemcpyAsync(agg, hcur, NH * sizeof(float), hipMemcpyDeviceToDevice, stream);
        scatter_add_kernel<<<edge_blocks, 256, 0, stream>>>(hcur, src, dst, agg, n_edges);

        // BN accumulators -> 0
        zero_kernel<<<1, 256, 0, stream>>>(stats, 256);

        // z = relu(agg@W1+b1)@W2 + b2 ; accumulate sum/sumsq per channel
        mlp_kernel<<<mlp_grid, 256, 0, stream>>>(
            agg, W1 + (size_t)l * H * H, b1 + (size_t)l * H,
            W2 + (size_t)l * H * H, b2 + (size_t)l * H,
            zbuf, stats, stats + 128, n_nodes);

        // fold BN into scale/shift, then h = relu(z*scale + shift)
        bn_finalize_kernel<<<1, H, 0, stream>>>(
            stats, stats + 128, gam + (size_t)l * H, bet + (size_t)l * H,
            stats + 256, stats + 384, 1.0f / (float)n_nodes);
        bn_apply_kernel<<<feat_blocks, 256, 0, stream>>>(
            zbuf, stats + 256, stats + 384, hbuf, feat4);

        hcur = hbuf;   // subsequent layers read/write d_out node region
    }

    // global mean pool
    zero_kernel<<<(64 * H + 255) / 256, 256, 0, stream>>>(gout, 64 * H);
    zero_kernel<<<1, 64, 0, stream>>>(counts, 64);
    count_kernel<<<(n_nodes + 255) / 256, 256, 0, stream>>>(bat, counts, n_nodes);
    pool_sum_kernel<<<feat_blocks, 256, 0, stream>>>(hbuf, bat, gout, feat4);
    pool_div_kernel<<<(64 * H + 255) / 256, 256, 0, stream>>>(gout, counts, 64 * H);
}